// LowRankGNNLayer_103079215396
// MI455X (gfx1250) — compile-verified
//
#include <hip/hip_runtime.h>

// ---------------------------------------------------------------------------
// LowRankGNN layer for MI455X (gfx1250, wave32).
// Roofline: edge aggregation (~3 GB irregular gather/atomic traffic, resolved
// mostly in the 192 MB L2) dominates; both GEMMs (~2 GFLOP) run on
// V_WMMA_F32_16X16X4_F32 in full fp32 with register-preloaded fragments.
// ---------------------------------------------------------------------------

typedef float v2f __attribute__((ext_vector_type(2)));
typedef float v8f __attribute__((ext_vector_type(8)));

#define NBR 4      // branches
#define DD  32     // per-branch dim
#define CC  128    // channels = NBR*DD
#define MCB 256    // codebook rows per branch

// ---------------------------------------------------------------------------
// Kernel 1: block-diagonal conv GEMM.
//   h[row][n*32+e] = sum_d Xin[row][n*32+d] * W_conv[n][d][e]
//   Xin row < B : X_B row ; row >= B : codebook[n][row-B][:] * warm_up_rate.
// One wave = one 16x16 tile, K=32 in 8 WMMA steps. Fragments preloaded, then
// back-to-back WMMAs. Store epilogue: wave-uniform full-tile fast path with
// immediate-offset store clause.
// WMMA fp32 layout: A lane -> row = lane&15, K = v + 2*(lane>>4);
//                   B lane -> col = lane&15, same K mapping;
//                   D vgpr r -> row = r + 8*(lane>>4), col = lane&15.
// ---------------------------------------------------------------------------
__global__ void conv_gemm_kernel(const float* __restrict__ X_B,
                                 const float* __restrict__ codebook,
                                 const float* __restrict__ W_conv,
                                 const float* __restrict__ wr_p,
                                 float* __restrict__ h,
                                 int Bn, int Mrows) {
  const int lane = threadIdx.x;          // 0..31, blockDim.x == 32
  const int half = lane >> 4;
  const int l16  = lane & 15;
  const int rowBase  = blockIdx.x * 16;
  const int n        = blockIdx.y >> 1;
  const int colHalf  = blockIdx.y & 1;
  const int c32      = colHalf * 16 + l16;   // output col within branch

  int rowA = rowBase + l16;
  if (rowA >= Mrows) rowA = Mrows - 1;       // clamp (exact for B+M = 50256)

  // Single pointer + scale selection (no per-step divergence; tiles never
  // straddle the B boundary because B % 16 == 0).
  const float* ap;
  float ascale;
  if (rowA < Bn) {
    ap = X_B + (size_t)rowA * CC + n * DD;
    ascale = 1.0f;
  } else {
    ap = codebook + ((size_t)n * MCB + (rowA - Bn)) * DD;
    ascale = wr_p[0];
  }
  const float* __restrict__ Wn = W_conv + (size_t)n * (DD * DD);

  // Preload all fragments (8 x v2f A, 8 x v2f B), then 8 back-to-back WMMAs.
  v2f Af[8], Bf[8];
#pragma unroll
  for (int i = 0; i < 8; ++i) {
    const int kA = 4 * i + 2 * half;
    Af[i].x = ap[kA]     * ascale;
    Af[i].y = ap[kA + 1] * ascale;
    Bf[i].x = Wn[kA * DD + c32];
    Bf[i].y = Wn[(kA + 1) * DD + c32];
  }
  v8f acc = {0.f, 0.f, 0.f, 0.f, 0.f, 0.f, 0.f, 0.f};
#pragma unroll
  for (int i = 0; i < 8; ++i)
    acc = __builtin_amdgcn_wmma_f32_16x16x4_f32(false, Af[i], false, Bf[i],
                                                (short)0, acc, false, false);

  // Store: one base pointer, immediate row offsets (rows differ by CC floats).
  const int col = n * DD + c32;
  float* hp = h + (size_t)(rowBase + 8 * half) * CC + col;
  if (rowBase + 16 <= Mrows) {               // wave-uniform fast path
#pragma unroll
    for (int r = 0; r < 8; ++r)
      hp[(size_t)r * CC] = acc[r];
  } else {
#pragma unroll
    for (int r = 0; r < 8; ++r)
      if (rowBase + r + 8 * half < Mrows) hp[(size_t)r * CC] = acc[r];
  }
}

// ---------------------------------------------------------------------------
// Kernel 2: initialize out_agg with broadcast b_conv (equivalent to adding
// the bias after aggregation) and zero the info_backward accumulator slot.
// ---------------------------------------------------------------------------
__global__ void init_kernel(const float* __restrict__ b_conv,
                            float* __restrict__ out_agg,
                            float* __restrict__ info_slot,
                            size_t total) {
  size_t i = (size_t)blockIdx.x * blockDim.x + threadIdx.x;
  if (i == 0) *info_slot = 0.f;
  const size_t stride = (size_t)gridDim.x * blockDim.x;
  for (; i < total; i += stride)
    out_agg[i] = b_conv[i & (CC - 1)];       // b_conv flat is exactly CC entries
}

// ---------------------------------------------------------------------------
// Kernel 3: edge-weighted aggregation. One wave per edge:
//   lane loads float4 (4 channels) of h[src], does 4 float atomics to out[dst].
// Edge scalars (src, dst, w) are wave-uniform: readfirstlane forces the index
// scalar so the compiler can fetch them through the SMEM path (s_load_b32,
// one request per wave) instead of 32-lane vector loads.
// remapSrc/remapDst route out-of-batch endpoints through c_indices -> B+m.
// ---------------------------------------------------------------------------
__global__ void edge_agg_kernel(float* __restrict__ out_agg,
                                const float* __restrict__ h,
                                const int* __restrict__ srcI,
                                const int* __restrict__ dstI,
                                const float* __restrict__ w,
                                const int* __restrict__ cidx,
                                int remapSrc, int remapDst,
                                int E, int Bn) {
  const int warp = (int)(((size_t)blockIdx.x * blockDim.x + threadIdx.x) >> 5);
  const int lane = threadIdx.x & 31;
  if (warp >= E) return;
  const int e = __builtin_amdgcn_readfirstlane(warp);   // wave-uniform index
  int s = srcI[e];
  int d = dstI[e];
  if (remapSrc) s = Bn + cidx[s];
  if (remapDst) d = Bn + cidx[d];
  const float wt = w[e];
  const float4 hv = *reinterpret_cast<const float4*>(h + (size_t)s * CC + lane * 4);
  float* op = out_agg + (size_t)d * CC + lane * 4;
  atomicAdd(op + 0, wt * hv.x);
  atomicAdd(op + 1, wt * hv.y);
  atomicAdd(op + 2, wt * hv.z);
  atomicAdd(op + 3, wt * hv.w);
}

// ---------------------------------------------------------------------------
// Kernel 4: info_backward = sum(out[:, B:] * vq_grad) * wr.
// 32768 elements; wave32 shuffle reduction, one atomic per wave.
// ---------------------------------------------------------------------------
__global__ void info_kernel(const float* __restrict__ out_agg,
                            const float* __restrict__ vq_grad,
                            const float* __restrict__ wr_p,
                            float* __restrict__ info_slot,
                            int Bn) {
  const int i = blockIdx.x * blockDim.x + threadIdx.x;  // 0..NBR*MCB*DD-1
  const int d = i & (DD - 1);
  const int m = (i >> 5) & (MCB - 1);
  const int n = i >> 13;
  float v = out_agg[(size_t)(Bn + m) * CC + n * DD + d] * vq_grad[i];
#pragma unroll
  for (int off = 16; off > 0; off >>= 1)
    v += __shfl_down(v, off, 32);
  if ((threadIdx.x & 31) == 0)
    atomicAdd(info_slot, v * wr_p[0]);
}

// ---------------------------------------------------------------------------
// Kernel 5: Y = out[:B] @ W_fc + b_fc + X_B.
// Block = 4 waves (128 threads). 128x16 W_fc column panel staged in LDS
// (8 KB) once per block, shared by 4 row tiles. K=128 processed in two
// chunks of 64: each chunk preloads 16 A-fragments (global v2f) and 16
// B-fragments (LDS) into registers, then runs 16 back-to-back WMMAs.
// LDS stride 16 is conflict-free (lanes 0-15 hit row K, 16-31 row K+2:
// disjoint bank groups). Epilogue uses immediate-offset loads/stores.
// ---------------------------------------------------------------------------
__global__ void fc_gemm_kernel(const float* __restrict__ Xcat,
                               const float* __restrict__ W_fc,
                               const float* __restrict__ b_fc,
                               const float* __restrict__ X_B,
                               float* __restrict__ Y,
                               int Bn) {
  __shared__ float Wl[CC * 16];
  const int tid = threadIdx.x;               // 0..127
  const int colBase = blockIdx.y * 16;

  // Stage W_fc[k][colBase..colBase+15] -> Wl[k*16 + j]; thread t owns row k=t.
  {
    const float* wp = W_fc + (size_t)tid * CC + colBase;
    float* lp = Wl + tid * 16;
#pragma unroll
    for (int j = 0; j < 16; ++j) lp[j] = wp[j];
  }
  __syncthreads();

  const int wave = tid >> 5;
  const int lane = tid & 31;
  const int half = lane >> 4;
  const int l16  = lane & 15;
  const int rowBase = blockIdx.x * 64 + wave * 16;
  if (rowBase >= Bn) return;                 // wave-uniform, after the barrier
  const int rowA = rowBase + l16;            // B % 16 == 0 -> always in range
  const float* __restrict__ arow = Xcat + (size_t)rowA * CC;

  v8f acc = {0.f, 0.f, 0.f, 0.f, 0.f, 0.f, 0.f, 0.f};
#pragma unroll
  for (int kc = 0; kc < 2; ++kc) {           // two K-chunks of 64
    v2f Af[16], Bf[16];
#pragma unroll
    for (int i = 0; i < 16; ++i) {
      const int kA = kc * 64 + 4 * i + 2 * half;   // even -> 8B-aligned v2f
      Af[i] = *reinterpret_cast<const v2f*>(arow + kA);
      Bf[i].x = Wl[kA * 16 + l16];
      Bf[i].y = Wl[(kA + 1) * 16 + l16];
    }
#pragma unroll
    for (int i = 0; i < 16; ++i)
      acc = __builtin_amdgcn_wmma_f32_16x16x4_f32(false, Af[i], false, Bf[i],
                                                  (short)0, acc, false, false);
  }

  // Epilogue: base pointers + immediate row offsets (rows differ by CC).
  const int col  = colBase + l16;
  const float bias = b_fc[col];
  const size_t base = (size_t)(rowBase + 8 * half) * CC + col;
  float* yp = Y + base;
  const float* xp = X_B + base;
#pragma unroll
  for (int r = 0; r < 8; ++r)
    yp[(size_t)r * CC] = acc[r] + bias + xp[(size_t)r * CC];
}

// ---------------------------------------------------------------------------
extern "C" void kernel_launch(void* const* d_in, const int* in_sizes, int n_in,
                              void* d_out, int out_size, void* d_ws, size_t ws_size,
                              hipStream_t stream) {
  const float* X_B      = (const float*)d_in[0];
  const int*   esrc_bb  = (const int*)  d_in[1];
  const int*   edst_bb  = (const int*)  d_in[2];
  const float* ew_bb    = (const float*)d_in[3];
  const int*   esrc_bm  = (const int*)  d_in[4];
  const int*   edst_bm  = (const int*)  d_in[5];
  const float* ew_bm    = (const float*)d_in[6];
  const int*   cidx     = (const int*)  d_in[7];
  const float* wr       = (const float*)d_in[8];
  const float* codebook = (const float*)d_in[9];
  const float* vq_grad  = (const float*)d_in[10];
  const float* W_conv   = (const float*)d_in[11];
  const float* b_conv   = (const float*)d_in[12];
  const float* W_fc     = (const float*)d_in[13];
  const float* b_fc     = (const float*)d_in[14];

  const int Bn    = in_sizes[0] / CC;     // 50000
  const int E_bb  = in_sizes[1];          // 1,000,000
  const int E_bm  = in_sizes[4];          // 1,000,000
  const int Mrows = Bn + MCB;             // 50256

  // Workspace: h then out_agg, each Mrows*CC floats (~25.7 MB each).
  float* h       = (float*)d_ws;
  float* out_agg = h + (size_t)Mrows * CC;

  float* Y         = (float*)d_out;
  float* info_slot = Y + (size_t)Bn * CC;   // scalar appended after Y

  // 1) conv GEMM -> h
  dim3 g1((Mrows + 15) / 16, NBR * 2);
  conv_gemm_kernel<<<g1, 32, 0, stream>>>(X_B, codebook, W_conv, wr, h, Bn, Mrows);

  // 2) out_agg = b_conv broadcast; info accumulator = 0
  init_kernel<<<2048, 256, 0, stream>>>(b_conv, out_agg,
                                        info_slot, (size_t)Mrows * CC);

  // 3) three edge-aggregation passes (BB, BM, MB), one wave per edge
  {
    int blk_bb = (int)(((size_t)E_bb * 32 + 255) / 256);
    edge_agg_kernel<<<blk_bb, 256, 0, stream>>>(out_agg, h, esrc_bb, edst_bb,
                                                ew_bb, cidx, 0, 0, E_bb, Bn);
    int blk_bm = (int)(((size_t)E_bm * 32 + 255) / 256);
    // A_BM: batch nodes receive from codeword rows (src remapped)
    edge_agg_kernel<<<blk_bm, 256, 0, stream>>>(out_agg, h, esrc_bm, edst_bm,
                                                ew_bm, cidx, 1, 0, E_bm, Bn);
    // A_MB: codeword rows receive from batch nodes (dst remapped)
    edge_agg_kernel<<<blk_bm, 256, 0, stream>>>(out_agg, h, edst_bm, esrc_bm,
                                                ew_bm, cidx, 0, 1, E_bm, Bn);
  }

  // 4) info_backward reduction over codeword rows
  info_kernel<<<(NBR * MCB * DD) / 256, 256, 0, stream>>>(out_agg, vq_grad, wr,
                                                          info_slot, Bn);

  // 5) FC GEMM + bias + skip -> Y
  dim3 g5((Bn + 63) / 64, CC / 16);
  fc_gemm_kernel<<<g5, 128, 0, stream>>>(out_agg, W_fc, b_fc, X_B, Y, Bn);
}